// ScatLayerj1_65566970740993
// MI455X (gfx1250) — compile-verified
//
#include <hip/hip_runtime.h>
#include <math.h>

#ifndef __has_builtin
#define __has_builtin(x) 0
#endif

#define NN 16
#define CC 64
#define HH 128
#define WW 128
#define H2 64
#define W2 64
#define SH 16                 // output rows per block
#define RROWS (2*SH + 6)      // 38 full-res rows incl. 3-row halo each side
#define LSTRIDE 128

// h1o tap values (symmetric)
#define H1_0 (-0.01071428571428571f)
#define H1_1 ( 0.05357142857142857f)
#define H1_2 ( 0.2607142857142857f)
#define H1_3 (-0.6071428571428571f)

typedef int v4i __attribute__((ext_vector_type(4)));
typedef __attribute__((address_space(1))) v4i* as1_v4i_ptr;
typedef __attribute__((address_space(3))) v4i* as3_v4i_ptr;

__device__ __forceinline__ int reflect128(int i) {
    // jnp.pad mode='symmetric' (half-sample), valid for -128 <= i <= 255
    i = i ^ (i >> 31);              // i<0 -> -1-i   (2 VALU, branchless)
    return (i > 127) ? 255 - i : i; // i>127 -> 255-i
}

__global__ __launch_bounds__(256)
void scat_dtcwt_fused(const float* __restrict__ x, float* __restrict__ out) {
    __shared__ float xs [RROWS * LSTRIDE];
    __shared__ float slo[RROWS * LSTRIDE];
    __shared__ float shi[RROWS * LSTRIDE];

    const int tid   = threadIdx.x;
    const int bid   = blockIdx.x;
    const int strip = bid & 3;       // 4 row-strips per plane
    const int plane = bid >> 2;      // n*CC + c
    const int i0    = strip * SH;    // output row base (half-res)
    const int y0    = 2 * i0;        // full-res row base
    const float* xp = x + (size_t)plane * (HH * WW);

    // ---------------- Phase 1: global -> LDS (CDNA5 async path) -------------
    const int total4 = RROWS * (WW / 4);   // 1216 float4 chunks
#if __has_builtin(__builtin_amdgcn_global_load_async_to_lds_b128)
    for (int f = tid; f < total4; f += 256) {
        const int row  = f >> 5;           // 32 float4 per row
        const int c4   = (f & 31) << 2;
        const int ysrc = reflect128(y0 - 3 + row);
        const float* g = xp + (size_t)ysrc * WW + c4;
        float*       l = xs + row * LSTRIDE + c4;
        __builtin_amdgcn_global_load_async_to_lds_b128(
            (as1_v4i_ptr)(void*)(const_cast<float*>(g)),
            (as3_v4i_ptr)(void*)l,
            0, 0);
    }
#if __has_builtin(__builtin_amdgcn_s_wait_asynccnt)
    __builtin_amdgcn_s_wait_asynccnt(0);
#else
    asm volatile("s_wait_asynccnt 0" ::: "memory");
#endif
#else
    for (int f = tid; f < total4; f += 256) {
        const int row  = f >> 5;
        const int c4   = (f & 31) << 2;
        const int ysrc = reflect128(y0 - 3 + row);
        const float4 v = *(const float4*)(xp + (size_t)ysrc * WW + c4);
        *(float4*)(xs + row * LSTRIDE + c4) = v;
    }
#endif
    __syncthreads();

    // -------- Phase 2: horizontal 5/7-tap, 4 columns per iteration ----------
    for (int f = tid; f < total4; f += 256) {
        const int row = f >> 5;
        const int c4  = (f & 31) << 2;
        const float* xr = xs + row * LSTRIDE;

        const float4 mid = *(const float4*)(xr + c4);   // cols c4..c4+3 (aligned)
        const float w3 = mid.x, w4 = mid.y, w5 = mid.z, w6 = mid.w;
        const float w0 = xr[reflect128(c4 - 3)];
        const float w1 = xr[reflect128(c4 - 2)];
        const float w2 = xr[reflect128(c4 - 1)];
        const float w7 = xr[reflect128(c4 + 4)];
        const float w8 = xr[reflect128(c4 + 5)];
        const float w9 = xr[reflect128(c4 + 6)];

        float4 lo, hi;
        {   // col c4   : window w0..w6, center w3
            const float sA = w2 + w4, sB = w1 + w5, sC = w0 + w6;
            lo.x = 0.6f*w3 + 0.25f*sA - 0.05f*sB;
            hi.x = H1_3*w3 + H1_2*sA + H1_1*sB + H1_0*sC;
        }
        {   // col c4+1 : window w1..w7, center w4
            const float sA = w3 + w5, sB = w2 + w6, sC = w1 + w7;
            lo.y = 0.6f*w4 + 0.25f*sA - 0.05f*sB;
            hi.y = H1_3*w4 + H1_2*sA + H1_1*sB + H1_0*sC;
        }
        {   // col c4+2 : window w2..w8, center w5
            const float sA = w4 + w6, sB = w3 + w7, sC = w2 + w8;
            lo.z = 0.6f*w5 + 0.25f*sA - 0.05f*sB;
            hi.z = H1_3*w5 + H1_2*sA + H1_1*sB + H1_0*sC;
        }
        {   // col c4+3 : window w3..w9, center w6
            const float sA = w5 + w7, sB = w4 + w8, sC = w3 + w9;
            lo.w = 0.6f*w6 + 0.25f*sA - 0.05f*sB;
            hi.w = H1_3*w6 + H1_2*sA + H1_1*sB + H1_0*sC;
        }
        *(float4*)(slo + row * LSTRIDE + c4) = lo;
        *(float4*)(shi + row * LSTRIDE + c4) = hi;
    }
    __syncthreads();

    // ---------------- Phase 3: vertical filters + q2c + magnitude ----------
    const float s = 0.70710678118654752f;  // 1/sqrt(2)
    const float bias  = 0.01f;
    const float bias2 = 0.0001f;
    const int j  = tid & 63;               // output column
    const int ib = tid >> 6;               // 0..3
    const int n  = plane >> 6;
    const int c  = plane & 63;
    const size_t kstride = (size_t)CC * H2 * W2;

#pragma unroll
    for (int q = 0; q < 4; ++q) {
        const int il = ib + 4 * q;         // local output row 0..15
        const int i  = i0 + il;            // global output row
        const int rb = 2 * il;             // local full-res row base (tap start)

        float ll[2][2], lh[2][2], hl[2][2], hh[2][2];
#pragma unroll
        for (int dy = 0; dy < 2; ++dy) {
#pragma unroll
            for (int dx = 0; dx < 2; ++dx) {
                const int off = (rb + dy) * LSTRIDE + (2*j + dx);
                const float* lp = slo + off;
                const float a0 = lp[0*LSTRIDE], a1 = lp[1*LSTRIDE],
                            a2 = lp[2*LSTRIDE], a3 = lp[3*LSTRIDE],
                            a4 = lp[4*LSTRIDE], a5 = lp[5*LSTRIDE],
                            a6 = lp[6*LSTRIDE];
                ll[dy][dx] = -0.05f*(a1 + a5) + 0.25f*(a2 + a4) + 0.6f*a3;
                lh[dy][dx] = H1_0*(a0 + a6) + H1_1*(a1 + a5) + H1_2*(a2 + a4) + H1_3*a3;
                const float* hp = shi + off;
                const float b0 = hp[0*LSTRIDE], b1 = hp[1*LSTRIDE],
                            b2 = hp[2*LSTRIDE], b3 = hp[3*LSTRIDE],
                            b4 = hp[4*LSTRIDE], b5 = hp[5*LSTRIDE],
                            b6 = hp[6*LSTRIDE];
                hl[dy][dx] = -0.05f*(b1 + b5) + 0.25f*(b2 + b4) + 0.6f*b3;
                hh[dy][dx] = H1_0*(b0 + b6) + H1_1*(b1 + b5) + H1_2*(b2 + b4) + H1_3*b3;
            }
        }

        // q2c: a=(0,0) b=(0,1) c=(1,0) d=(1,1), each scaled by 1/sqrt(2)
        // lh -> d15, d165 ; hh -> d45, d135 ; hl -> d75, d105
        const float la = s*lh[0][0], lb = s*lh[0][1], lc = s*lh[1][0], ld = s*lh[1][1];
        const float ha = s*hh[0][0], hb = s*hh[0][1], hc = s*hh[1][0], hd = s*hh[1][1];
        const float ga = s*hl[0][0], gb = s*hl[0][1], gc = s*hl[1][0], gd = s*hl[1][1];

        const float r15  = la - ld, i15  = lb + lc;
        const float r165 = la + ld, i165 = lb - lc;
        const float r45  = ha - hd, i45  = hb + hc;
        const float r135 = ha + hd, i135 = hb - hc;
        const float r75  = ga - gd, i75  = gb + gc;
        const float r105 = ga + gd, i105 = gb - gc;

        const float m15  = sqrtf(r15*r15   + i15*i15   + bias2) - bias;
        const float m45  = sqrtf(r45*r45   + i45*i45   + bias2) - bias;
        const float m75  = sqrtf(r75*r75   + i75*i75   + bias2) - bias;
        const float m105 = sqrtf(r105*r105 + i105*i105 + bias2) - bias;
        const float m135 = sqrtf(r135*r135 + i135*i135 + bias2) - bias;
        const float m165 = sqrtf(r165*r165 + i165*i165 + bias2) - bias;

        const float ll2 = 0.25f * (ll[0][0] + ll[0][1] + ll[1][0] + ll[1][1]);

        // out[n][k][c][i][j], dims (N,7,C,H2,W2)
        const size_t base = ((((size_t)n * 7) * CC + c) * H2 + i) * W2 + j;
        out[base + 0*kstride] = ll2;
        out[base + 1*kstride] = m15;
        out[base + 2*kstride] = m45;
        out[base + 3*kstride] = m75;
        out[base + 4*kstride] = m105;
        out[base + 5*kstride] = m135;
        out[base + 6*kstride] = m165;
    }
}

extern "C" void kernel_launch(void* const* d_in, const int* in_sizes, int n_in,
                              void* d_out, int out_size, void* d_ws, size_t ws_size,
                              hipStream_t stream) {
    const float* x = (const float*)d_in[0];
    float* out = (float*)d_out;
    // 16 n * 64 channels * 4 strips = 4096 blocks
    dim3 grid(NN * CC * 4);
    dim3 block(256);
    hipLaunchKernelGGL(scat_dtcwt_fused, grid, block, 0, stream, x, out);
}